// MultiHeadAttention_68659347194437
// MI455X (gfx1250) — compile-verified
//
#include <hip/hip_runtime.h>

typedef float v2f __attribute__((ext_vector_type(2)));
typedef float v4f __attribute__((ext_vector_type(4)));
typedef float v8f __attribute__((ext_vector_type(8)));

#define CCH 32
#define HH  64
#define WW  512
#define RS  (HH*WW)        /* 32768 floats: stride between channel planes */
#define BSTR (CCH*RS)      /* per-batch stride in floats */

__global__ __launch_bounds__(256)
void linattn_kernel(const float* __restrict__ q, const float* __restrict__ k,
                    const float* __restrict__ v,
                    const float* __restrict__ wq, const float* __restrict__ bq,
                    const float* __restrict__ wk, const float* __restrict__ bk,
                    const float* __restrict__ wv, const float* __restrict__ bv,
                    float* __restrict__ out)
{
    __shared__ float sWq[CCH*CCH], sWk[CCH*CCH], sWv[CCH*CCH];
    __shared__ float sBq[CCH], sBk[CCH], sBv[CCH];
    __shared__ float sG[CCH*CCH];     // K_raw V_raw^T
    __shared__ float sT1[CCH*CCH];    // G * Wv^T
    __shared__ float sM[CCH*CCH];     // kp^T vp
    __shared__ float sP[CCH*CCH];     // M^T Wq + I
    __shared__ float sks[CCH], svs[CCH], swksk[CCH], swvsv[CCH], sR[CCH];

    const int tid = threadIdx.x;
    const int bh  = blockIdx.x;          // 0..511  == b*64 + h
    const int b   = bh >> 6;
    const int h   = bh & 63;
    const long base = (long)b * BSTR + (long)h * WW;
    const float* qb = q + base;
    const float* kb = k + base;
    const float* vb = v + base;
    float*       ob = out + base;

    // stage weights into LDS, zero the reduction buffers
    for (int i = tid; i < CCH*CCH; i += 256) {
        sWq[i] = wq[i]; sWk[i] = wk[i]; sWv[i] = wv[i];
        sG[i] = 0.0f;
    }
    if (tid < CCH) {
        sBq[tid] = bq[tid]; sBk[tid] = bk[tid]; sBv[tid] = bv[tid];
        sks[tid] = 0.0f; svs[tid] = 0.0f;
    }
    __syncthreads();

    const int lane = tid & 31;
    const int wvid = tid >> 5;     // wave 0..7, owns w-chunk wvid*64
    const int lr   = lane & 15;    // row/col within tile
    const int lh   = lane >> 4;    // lane half selects K-slot pair

    // ---------------- Phase A: G = K V^T, sk, sv (WMMA f32 16x16x4) --------
    // b128 per lane at w0 + lh*4: .xy feeds K-step t, .zw feeds K-step t+1.
    // The w->K-slot mapping is permuted but identical for A(k) and B(v),
    // so the contraction over w is exact.
    {
        v8f a00 = {}, a01 = {}, a10 = {}, a11 = {};
        float ks0 = 0.f, ks1 = 0.f, vs0 = 0.f, vs1 = 0.f;
        const int wc = wvid * 64;
        const float* k0p = kb + (long)lr * RS + wc + lh * 4;
        const float* k1p = k0p + 16L * RS;
        const float* v0p = vb + (long)lr * RS + wc + lh * 4;
        const float* v1p = v0p + 16L * RS;
        for (int t = 0; t < 8; ++t) {
            const int off = t * 8;
            v4f ka0 = *(const v4f*)(k0p + off);   // A: k rows 0..15
            v4f ka1 = *(const v4f*)(k1p + off);   // A: k rows 16..31
            v4f vb0 = *(const v4f*)(v0p + off);   // B: v rows 0..15
            v4f vb1 = *(const v4f*)(v1p + off);   // B: v rows 16..31
            ks0 += ka0.x + ka0.y + ka0.z + ka0.w;
            ks1 += ka1.x + ka1.y + ka1.z + ka1.w;
            vs0 += vb0.x + vb0.y + vb0.z + vb0.w;
            vs1 += vb1.x + vb1.y + vb1.z + vb1.w;
            v2f aL0 = {ka0.x, ka0.y}, aH0 = {ka0.z, ka0.w};
            v2f aL1 = {ka1.x, ka1.y}, aH1 = {ka1.z, ka1.w};
            v2f bL0 = {vb0.x, vb0.y}, bH0 = {vb0.z, vb0.w};
            v2f bL1 = {vb1.x, vb1.y}, bH1 = {vb1.z, vb1.w};
            a00 = __builtin_amdgcn_wmma_f32_16x16x4_f32(false, aL0, false, bL0, (short)0, a00, false, false);
            a01 = __builtin_amdgcn_wmma_f32_16x16x4_f32(false, aL0, false, bL1, (short)0, a01, false, false);
            a10 = __builtin_amdgcn_wmma_f32_16x16x4_f32(false, aL1, false, bL0, (short)0, a10, false, false);
            a11 = __builtin_amdgcn_wmma_f32_16x16x4_f32(false, aL1, false, bL1, (short)0, a11, false, false);
            a00 = __builtin_amdgcn_wmma_f32_16x16x4_f32(false, aH0, false, bH0, (short)0, a00, false, false);
            a01 = __builtin_amdgcn_wmma_f32_16x16x4_f32(false, aH0, false, bH1, (short)0, a01, false, false);
            a10 = __builtin_amdgcn_wmma_f32_16x16x4_f32(false, aH1, false, bH0, (short)0, a10, false, false);
            a11 = __builtin_amdgcn_wmma_f32_16x16x4_f32(false, aH1, false, bH1, (short)0, a11, false, false);
        }
        atomicAdd(&sks[lr],      ks0);
        atomicAdd(&sks[16 + lr], ks1);
        atomicAdd(&svs[lr],      vs0);
        atomicAdd(&svs[16 + lr], vs1);
        #pragma unroll
        for (int i = 0; i < 8; ++i) {
            const int er = i + lh * 8;            // C/D layout row
            atomicAdd(&sG[er * CCH + lr],             a00[i]);
            atomicAdd(&sG[er * CCH + 16 + lr],        a01[i]);
            atomicAdd(&sG[(16 + er) * CCH + lr],      a10[i]);
            atomicAdd(&sG[(16 + er) * CCH + 16 + lr], a11[i]);
        }
    }
    __syncthreads();

    // ---------------- Phase B: tiny 32x32 algebra --------------------------
    // wksk[d] = (Wk sk)[d];  wvsv[c] = (Wv sv)[c]
    if (tid < 64) {
        const int d = tid & 31;
        float s = 0.f;
        if (tid < 32) { for (int e = 0; e < 32; ++e) s += sWk[d*32+e] * sks[e]; swksk[d] = s; }
        else          { for (int f = 0; f < 32; ++f) s += sWv[d*32+f] * svs[f]; swvsv[d] = s; }
    }
    // T1[e,c] = sum_f G[e,f] * Wv[c,f]
    #pragma unroll
    for (int j = 0; j < 4; ++j) {
        const int o = tid + 256*j, e = o >> 5, c = o & 31;
        float s = 0.f;
        for (int f = 0; f < 32; ++f) s += sG[e*32+f] * sWv[c*32+f];
        sT1[e*32+c] = s;
    }
    __syncthreads();
    // M[d,c] = sum_e Wk[d,e] T1[e,c] + bk[d]*wvsv[c] + bv[c]*wksk[d] + 512*bk[d]*bv[c]
    #pragma unroll
    for (int j = 0; j < 4; ++j) {
        const int o = tid + 256*j, d = o >> 5, c = o & 31;
        float s = 0.f;
        for (int e = 0; e < 32; ++e) s += sWk[d*32+e] * sT1[e*32+c];
        s += sBk[d]*swvsv[c] + sBv[c]*swksk[d] + 512.0f*sBk[d]*sBv[c];
        sM[d*32+c] = s;
    }
    __syncthreads();
    // P[c,e] = sum_d M[d,c] Wq[d,e]  (+1 on diag folds in the residual)
    #pragma unroll
    for (int j = 0; j < 4; ++j) {
        const int o = tid + 256*j, c = o >> 5, e = o & 31;
        float s = 0.f;
        for (int d = 0; d < 32; ++d) s += sM[d*32+c] * sWq[d*32+e];
        if (c == e) s += 1.0f;
        sP[c*32+e] = s;
    }
    if (tid < 32) {       // r[c] = sum_d M[d,c] bq[d]
        float s = 0.f;
        for (int d = 0; d < 32; ++d) s += sM[d*32+tid] * sBq[d];
        sR[tid] = s;
    }
    __syncthreads();

    // ---------------- Phase C: out = P q + r  (WMMA f32 16x16x4) -----------
    {
        v2f pa[2][8];
        #pragma unroll
        for (int mt = 0; mt < 2; ++mt)
            #pragma unroll
            for (int ks2 = 0; ks2 < 8; ++ks2) {
                const int m  = mt*16 + lr;
                const int kk = ks2*4 + lh*2;
                pa[mt][ks2] = *(const v2f*)(&sP[m*32 + kk]);
            }
        float rinit[2][8];
        #pragma unroll
        for (int mt = 0; mt < 2; ++mt)
            #pragma unroll
            for (int i = 0; i < 8; ++i)
                rinit[mt][i] = sR[mt*16 + i + lh*8];

        const int wc = wvid * 64;
        for (int nt = 0; nt < 4; ++nt) {
            const int w0 = wc + nt*16;
            v8f acc0, acc1;
            #pragma unroll
            for (int i = 0; i < 8; ++i) { acc0[i] = rinit[0][i]; acc1[i] = rinit[1][i]; }
            const float* qp0 = qb + (long)(lh*2)*RS + w0 + lr;
            #pragma unroll
            for (int ks2 = 0; ks2 < 8; ++ks2) {
                const float* p = qp0 + (long)(ks2*4)*RS;
                v2f bf;                 // B frag: rows k, k+1 at column w0+lr
                bf.x = p[0];
                bf.y = p[RS];
                acc0 = __builtin_amdgcn_wmma_f32_16x16x4_f32(false, pa[0][ks2], false, bf, (short)0, acc0, false, false);
                acc1 = __builtin_amdgcn_wmma_f32_16x16x4_f32(false, pa[1][ks2], false, bf, (short)0, acc1, false, false);
            }
            #pragma unroll
            for (int i = 0; i < 8; ++i) {
                const int c0 = i + lh*8;
                ob[(long)c0 * RS      + w0 + lr] = acc0[i];
                ob[(long)(16+c0) * RS + w0 + lr] = acc1[i];
            }
        }
    }
}

extern "C" void kernel_launch(void* const* d_in, const int* in_sizes, int n_in,
                              void* d_out, int out_size, void* d_ws, size_t ws_size,
                              hipStream_t stream) {
    (void)in_sizes; (void)n_in; (void)out_size; (void)d_ws; (void)ws_size;
    const float* q  = (const float*)d_in[0];
    const float* k  = (const float*)d_in[1];
    const float* v  = (const float*)d_in[2];
    const float* wq = (const float*)d_in[3];
    const float* bq = (const float*)d_in[4];
    const float* wk = (const float*)d_in[5];
    const float* bk = (const float*)d_in[6];
    const float* wv = (const float*)d_in[7];
    const float* bv = (const float*)d_in[8];
    float* out = (float*)d_out;
    dim3 grid(8 * 64);   // one block per (b,h)
    dim3 block(256);     // 8 wave32
    linattn_kernel<<<grid, block, 0, stream>>>(q, k, v, wq, bq, wk, bk, wv, bv, out);
}